// GCN_37941741093409
// MI455X (gfx1250) — compile-verified
//
#include <hip/hip_runtime.h>
#include <hip/hip_bf16.h>

#define N_NODES  50000
#define N_EDGES  800000
#define N_GRAPHS 128
#define IN_DIM   256
#define OUT_DIM  128
#define NEG_SLOPE 0.01f

#define TILE_M   80          // rows per block (50000 = 625 * 80)
#define NT       5           // 16-row WMMA tiles per block
#define LDS_S    81          // padded m-stride (odd -> bank-conflict-free frag reads)

typedef float v2f __attribute__((ext_vector_type(2)));
typedef float v8f __attribute__((ext_vector_type(8)));

// ---------------------------------------------------------------- zero fill
__global__ void GCN_zero_kernel(float* __restrict__ p, size_t n) {
    size_t i = (size_t)blockIdx.x * blockDim.x + threadIdx.x;
    size_t stride = (size_t)gridDim.x * blockDim.x;
    for (; i < n; i += stride) p[i] = 0.0f;
}

// ---------------------------------------------------------------- degrees
__global__ __launch_bounds__(256) void GCN_deg_kernel(
    const int* __restrict__ src, const int* __restrict__ dst,
    float* __restrict__ out_deg, float* __restrict__ in_deg, int n_edges) {
    int e = blockIdx.x * blockDim.x + threadIdx.x;
    if (e < n_edges) {
        unsafeAtomicAdd(&out_deg[src[e]], 1.0f);
        unsafeAtomicAdd(&in_deg[dst[e]], 1.0f);
    }
}

// ---------------------------------------------------------------- xs = (feat @ w_red + b_red) * out_norm
// Block = 80-row M tile (5 WMMA tiles), 8 waves each own one 16-wide N slice.
// Each B fragment load is reused by 5 independent WMMA accumulator chains.
__global__ __launch_bounds__(256) void GCN_h0_gemm_kernel(
    const float* __restrict__ feat, const float* __restrict__ w_red,
    const float* __restrict__ b_red, const float* __restrict__ out_deg,
    float* __restrict__ xs) {
    __shared__ float ldsA[IN_DIM * LDS_S];   // A transposed, k-major
    __shared__ float ldsN[TILE_M];
    const int t = threadIdx.x;
    const int row0 = blockIdx.x * TILE_M;

    if (t < TILE_M) {
        float d = out_deg[row0 + t];
        ldsN[t] = rsqrtf(d > 1.0f ? d : 1.0f);
    }
    // Stage 80x256 fp32 A tile: coalesced float4 reads, transposed scatter to LDS.
#pragma unroll
    for (int it = 0; it < (TILE_M * IN_DIM / 4) / 256; ++it) {   // 20 iters
        int f = it * 256 + t;
        int r = f >> 6;                 // 64 float4 per row
        int k = (f & 63) * 4;
        float4 v = *(const float4*)(feat + (size_t)(row0 + r) * IN_DIM + k);
        ldsA[(k + 0) * LDS_S + r] = v.x;
        ldsA[(k + 1) * LDS_S + r] = v.y;
        ldsA[(k + 2) * LDS_S + r] = v.z;
        ldsA[(k + 3) * LDS_S + r] = v.w;
    }
    __syncthreads();

    const int lane  = t & 31;
    const int wave  = t >> 5;
    const int m     = lane & 15;
    const int khalf = (lane >> 4) * 2;    // lanes 16..31 hold k+2,k+3
    const int n     = wave * 16 + m;      // output column

    v8f acc[NT] = {};
    for (int k = 0; k < IN_DIM; k += 4) {
        const int ka = k + khalf;
        v2f b;
        b.x = w_red[(size_t)(ka + 0) * OUT_DIM + n];
        b.y = w_red[(size_t)(ka + 1) * OUT_DIM + n];
#pragma unroll
        for (int tile = 0; tile < NT; ++tile) {
            v2f a;
            a.x = ldsA[(ka + 0) * LDS_S + tile * 16 + m];
            a.y = ldsA[(ka + 1) * LDS_S + tile * 16 + m];
            acc[tile] = __builtin_amdgcn_wmma_f32_16x16x4_f32(
                false, a, false, b, (short)0, acc[tile], false, false);
        }
    }

    const float bn = b_red[n];
    const int half8 = (lane >> 4) << 3;   // C/D layout: vgpr r -> M=r / r+8
#pragma unroll
    for (int tile = 0; tile < NT; ++tile) {
#pragma unroll
        for (int r = 0; r < 8; ++r) {
            int lr = tile * 16 + r + half8;
            xs[(size_t)(row0 + lr) * OUT_DIM + n] = (acc[tile][r] + bn) * ldsN[lr];
        }
    }
}

// ---------------------------------------------------------------- agg[dst] += xs[src] (one wave per edge)
__global__ __launch_bounds__(256) void GCN_edge_kernel(
    const int* __restrict__ src, const int* __restrict__ dst,
    const float* __restrict__ xs, float* __restrict__ agg, int n_edges) {
    int e = blockIdx.x * 8 + (threadIdx.x >> 5);
    if (e >= n_edges) return;
    const int lane = threadIdx.x & 31;
    const int s = src[e];
    const int d = dst[e];
    float4 v = *(const float4*)(xs + (size_t)s * OUT_DIM + lane * 4);
    float* a = agg + (size_t)d * OUT_DIM + lane * 4;
    unsafeAtomicAdd(a + 0, v.x);
    unsafeAtomicAdd(a + 1, v.y);
    unsafeAtomicAdd(a + 2, v.z);
    unsafeAtomicAdd(a + 3, v.w);
}

// ---------------------------------------------------------------- out = leaky_relu((agg*in_norm) @ w + b); g[gid] += out
__global__ __launch_bounds__(256) void GCN_gcn_gemm_kernel(
    const float* __restrict__ agg, const float* __restrict__ w,
    const float* __restrict__ bias, const float* __restrict__ in_deg,
    const int* __restrict__ graph_ids, float* __restrict__ g) {
    __shared__ float ldsA[OUT_DIM * LDS_S];
    __shared__ float ldsN[TILE_M];
    __shared__ int   ldsG[TILE_M];
    const int t = threadIdx.x;
    const int row0 = blockIdx.x * TILE_M;

    if (t < TILE_M) {
        float d = in_deg[row0 + t];
        ldsN[t] = rsqrtf(d > 1.0f ? d : 1.0f);
        ldsG[t] = graph_ids[row0 + t];
    }
    __syncthreads();   // staging reads ldsN

#pragma unroll
    for (int it = 0; it < (TILE_M * OUT_DIM / 4) / 256; ++it) {  // 10 iters
        int f = it * 256 + t;
        int r = f >> 5;                 // 32 float4 per row
        int k = (f & 31) * 4;
        float nrm = ldsN[r];
        float4 v = *(const float4*)(agg + (size_t)(row0 + r) * OUT_DIM + k);
        ldsA[(k + 0) * LDS_S + r] = v.x * nrm;
        ldsA[(k + 1) * LDS_S + r] = v.y * nrm;
        ldsA[(k + 2) * LDS_S + r] = v.z * nrm;
        ldsA[(k + 3) * LDS_S + r] = v.w * nrm;
    }
    __syncthreads();

    const int lane  = t & 31;
    const int wave  = t >> 5;
    const int m     = lane & 15;
    const int khalf = (lane >> 4) * 2;
    const int n     = wave * 16 + m;

    v8f acc[NT] = {};
    for (int k = 0; k < OUT_DIM; k += 4) {
        const int ka = k + khalf;
        v2f b;
        b.x = w[(size_t)(ka + 0) * OUT_DIM + n];
        b.y = w[(size_t)(ka + 1) * OUT_DIM + n];
#pragma unroll
        for (int tile = 0; tile < NT; ++tile) {
            v2f a;
            a.x = ldsA[(ka + 0) * LDS_S + tile * 16 + m];
            a.y = ldsA[(ka + 1) * LDS_S + tile * 16 + m];
            acc[tile] = __builtin_amdgcn_wmma_f32_16x16x4_f32(
                false, a, false, b, (short)0, acc[tile], false, false);
        }
    }

    const float bn = bias[n];
    const int half8 = (lane >> 4) << 3;
#pragma unroll
    for (int tile = 0; tile < NT; ++tile) {
#pragma unroll
        for (int r = 0; r < 8; ++r) {
            int lr = tile * 16 + r + half8;
            float v = acc[tile][r] + bn;
            v = v > 0.0f ? v : v * NEG_SLOPE;
            unsafeAtomicAdd(&g[(size_t)ldsG[lr] * OUT_DIM + n], v);
        }
    }
}

// ---------------------------------------------------------------- result = g @ w_cls + b_cls  [128 x 2]
__global__ __launch_bounds__(256) void GCN_cls_kernel(
    const float* __restrict__ g, const float* __restrict__ w_cls,
    const float* __restrict__ b_cls, float* __restrict__ out) {
    const int t = threadIdx.x;          // 256 threads = 128 graphs x 2 cols
    const int gr = t >> 1;
    const int c  = t & 1;
    float s = b_cls[c];
    const float* row = g + (size_t)gr * OUT_DIM;
    for (int k = 0; k < OUT_DIM; ++k) s += row[k] * w_cls[k * 2 + c];
    out[gr * 2 + c] = s;
}

// ---------------------------------------------------------------- launcher
extern "C" void kernel_launch(void* const* d_in, const int* in_sizes, int n_in,
                              void* d_out, int out_size, void* d_ws, size_t ws_size,
                              hipStream_t stream) {
    const float* feat   = (const float*)d_in[0];
    const int*   src    = (const int*)d_in[1];
    const int*   dst    = (const int*)d_in[2];
    const int*   gids   = (const int*)d_in[3];
    const float* w_red  = (const float*)d_in[4];
    const float* b_red  = (const float*)d_in[5];
    const float* w_gcn  = (const float*)d_in[6];
    const float* b_gcn  = (const float*)d_in[7];
    const float* w_cls  = (const float*)d_in[8];
    const float* b_cls  = (const float*)d_in[9];

    float* ws      = (float*)d_ws;
    float* out_deg = ws;                         // 50000
    float* in_deg  = ws + 50000;                 // 50000
    float* agg     = ws + 100000;                // 50000*128
    float* g       = ws + 100000 + (size_t)N_NODES * OUT_DIM;          // 128*128
    float* xs      = g + (size_t)N_GRAPHS * OUT_DIM;                   // 50000*128
    const size_t n_zero = 100000 + (size_t)N_NODES * OUT_DIM + (size_t)N_GRAPHS * OUT_DIM;

    GCN_zero_kernel<<<4096, 256, 0, stream>>>(ws, n_zero);
    GCN_deg_kernel<<<(N_EDGES + 255) / 256, 256, 0, stream>>>(src, dst, out_deg, in_deg, N_EDGES);
    GCN_h0_gemm_kernel<<<N_NODES / TILE_M, 256, 0, stream>>>(feat, w_red, b_red, out_deg, xs);
    GCN_edge_kernel<<<(N_EDGES + 7) / 8, 256, 0, stream>>>(src, dst, xs, agg, N_EDGES);
    // Only the last GCN layer's activation survives the reference loop -> layer index 2.
    GCN_gcn_gemm_kernel<<<N_NODES / TILE_M, 256, 0, stream>>>(
        agg, w_gcn + 2 * OUT_DIM * OUT_DIM, b_gcn + 2 * OUT_DIM, in_deg, gids, g);
    GCN_cls_kernel<<<1, 256, 0, stream>>>(g, w_cls, b_cls, (float*)d_out);
}